// GCN_binary_56487409877510
// MI455X (gfx1250) — compile-verified
//
#include <hip/hip_runtime.h>

typedef __attribute__((ext_vector_type(2))) float v2f;
typedef __attribute__((ext_vector_type(8))) float v8f;

#define N_NODES 100000
#define N_EDGES 1600000
#define NFEAT   256
#define NHID    128

// ---------------------------------------------------------------- utilities
__global__ void zero_kernel(float* __restrict__ p, int n) {
    int i = blockIdx.x * blockDim.x + threadIdx.x;
    if (i < n) p[i] = 0.0f;
}

// degrees via float atomics into (zeroed) degS/degD
__global__ void degree_kernel(const int* __restrict__ src, const int* __restrict__ dst,
                              float* __restrict__ degS, float* __restrict__ degD, int e) {
    int i = blockIdx.x * blockDim.x + threadIdx.x;
    if (i < e) {
        atomicAdd(&degS[src[i]], 1.0f);
        atomicAdd(&degD[dst[i]], 1.0f);
    }
}

// deg -> max(deg,1)^-0.5 in place
__global__ void norm_finalize(float* __restrict__ degS, float* __restrict__ degD, int n) {
    int i = blockIdx.x * blockDim.x + threadIdx.x;
    if (i < n) {
        degS[i] = rsqrtf(fmaxf(degS[i], 1.0f));
        degD[i] = rsqrtf(fmaxf(degD[i], 1.0f));
    }
}

// ------------------------------------------------- WMMA fp32 GEMM (row-scaled)
// Out[N_rows x 128] = (X * scale[:,None]) @ W,  X: [N_rows x K], W: [K x 128]
// Block = 256 threads = 8 waves. Block covers 16 rows; wave w covers cols
// [16w, 16w+16). K stepped by 4 with V_WMMA_F32_16X16X4_F32.
//
// A-frag (16x4 f32, ISA 7.12.2): lanes l=lane%16 hold M=l; half=lane/16:
//   v[0] = A[M][2*half], v[1] = A[M][2*half+1]
// B-frag (4x16) mirrors A; C/D: vgpr r -> M=r+8*half, N=lane%16.
__global__ void __launch_bounds__(256)
gemm_scaled_wmma(const float* __restrict__ X, const float* __restrict__ scale,
                 const float* __restrict__ W, float* __restrict__ Out, int K) {
    const int wave = threadIdx.x >> 5;
    const int lane = threadIdx.x & 31;
    const int kh   = lane >> 4;     // half: 0 or 1
    const int l    = lane & 15;
    const int row0 = blockIdx.x << 4;      // 16 rows per block (100000 = 6250*16)
    const int col0 = wave << 4;            // 8 waves * 16 = 128 cols
    const int row  = row0 + l;

    const float ns = scale[row];
    const float* __restrict__ xrow = X + (size_t)row * K;

    v8f c = {};
    for (int k = 0; k < K; k += 4) {
        const int ka = k + 2 * kh;
        v2f a;
        a.x = xrow[ka]     * ns;
        a.y = xrow[ka + 1] * ns;
        v2f b;
        b.x = W[(size_t)ka       * NHID + col0 + l];
        b.y = W[(size_t)(ka + 1) * NHID + col0 + l];
        // 8 args: (neg_a, A, neg_b, B, c_mod, C, reuse_a, reuse_b)
        c = __builtin_amdgcn_wmma_f32_16x16x4_f32(false, a, false, b,
                                                  (short)0, c, false, false);
    }

    float* __restrict__ orow = Out + (size_t)row0 * NHID + col0;
#pragma unroll
    for (int r = 0; r < 8; ++r) {
        orow[(size_t)(r + 8 * kh) * NHID + l] = c[r];
    }
}

// --------------------------------------------------- edge scatter (SpMM part)
// one wave per edge: gather 128 floats of M[src] (float4/lane, 512B contiguous
// per wave) and atomically add into AGG[dst].
__global__ void __launch_bounds__(256)
scatter_add(const float* __restrict__ M, const int* __restrict__ src,
            const int* __restrict__ dst, float* __restrict__ AGG, int e) {
    const int w    = (blockIdx.x * blockDim.x + threadIdx.x) >> 5;
    const int lane = threadIdx.x & 31;
    if (w >= e) return;
    const int s = src[w];
    const int d = dst[w];
    const float4 v = reinterpret_cast<const float4*>(M + (size_t)s * NHID)[lane];
    float* drow = AGG + (size_t)d * NHID + lane * 4;
    atomicAdd(drow + 0, v.x);
    atomicAdd(drow + 1, v.y);
    atomicAdd(drow + 2, v.z);
    atomicAdd(drow + 3, v.w);
}

// H = relu(AGG * normD[row] + b)
__global__ void bias_relu(const float* __restrict__ AGG, const float* __restrict__ normD,
                          const float* __restrict__ b, float* __restrict__ H, int total) {
    int i = blockIdx.x * blockDim.x + threadIdx.x;
    if (i < total) {
        const int row = i >> 7;       // /128
        const int j   = i & 127;
        const float v = AGG[i] * normD[row] + b[j];
        H[i] = v > 0.0f ? v : 0.0f;
    }
}

// out[i] = sum_j (AGG[i][j]*normD[i] + b2[j]) * Wfc[j] + bfc   (wave per node)
__global__ void __launch_bounds__(256)
final_fc(const float* __restrict__ AGG, const float* __restrict__ normD,
         const float* __restrict__ b2, const float* __restrict__ Wfc,
         const float* __restrict__ bfc, float* __restrict__ out, int n) {
    const int w    = (blockIdx.x * blockDim.x + threadIdx.x) >> 5;
    const int lane = threadIdx.x & 31;
    if (w >= n) return;
    const float nd = normD[w];
    const float* arow = AGG + (size_t)w * NHID;
    float acc = 0.0f;
#pragma unroll
    for (int t = 0; t < 4; ++t) {
        const int j = lane + 32 * t;
        acc += (arow[j] * nd + b2[j]) * Wfc[j];
    }
#pragma unroll
    for (int off = 16; off > 0; off >>= 1)
        acc += __shfl_xor(acc, off, 32);
    if (lane == 0) out[w] = acc + bfc[0];
}

// --------------------------------------------------------------------- launch
extern "C" void kernel_launch(void* const* d_in, const int* in_sizes, int n_in,
                              void* d_out, int out_size, void* d_ws, size_t ws_size,
                              hipStream_t stream) {
    const float* x   = (const float*)d_in[0];   // [N, 256]
    const float* W1  = (const float*)d_in[1];   // [256, 128]
    const float* b1  = (const float*)d_in[2];   // [128]
    const float* W2  = (const float*)d_in[3];   // [128, 128]
    const float* b2  = (const float*)d_in[4];   // [128]
    const float* Wfc = (const float*)d_in[5];   // [128, 1]
    const float* bfc = (const float*)d_in[6];   // [1]
    const int*   src = (const int*)d_in[7];     // [E]
    const int*   dst = (const int*)d_in[8];     // [E]
    float* out = (float*)d_out;                 // [N, 1]

    const int N = N_NODES, E = N_EDGES;
    const int NH = NHID;

    // workspace: BUF_A [N*128], BUF_B [N*128], normS [N], normD [N]
    float* BUF_A = (float*)d_ws;
    float* BUF_B = BUF_A + (size_t)N * NH;
    float* normS = BUF_B + (size_t)N * NH;
    float* normD = normS + N;

    const int TPB = 256;
    const int featTotal = N * NH;                 // 12.8M
    const int gridFeat  = (featTotal + TPB - 1) / TPB;
    const int gridN     = (N + TPB - 1) / TPB;
    const int gridE     = (E + TPB - 1) / TPB;
    const int gridEdgeWaves = (E * (TPB / (TPB / 32)) ) ? ((E + 7) / 8) : 0; // 8 waves/block
    const int gridNodeWaves = (N + 7) / 8;
    const int gridGemm  = N / 16;                 // 6250 exactly

    // degrees -> norms (normS/normD contiguous: zero both in one shot)
    zero_kernel<<<(2 * N + TPB - 1) / TPB, TPB, 0, stream>>>(normS, 2 * N);
    degree_kernel<<<gridE, TPB, 0, stream>>>(src, dst, normS, normD, E);
    norm_finalize<<<gridN, TPB, 0, stream>>>(normS, normD, N);

    // layer 1: M1 = (x*normS)@W1 ; AGG1 = scatter ; H1 = relu(AGG1*normD + b1)
    zero_kernel<<<gridFeat, TPB, 0, stream>>>(BUF_B, featTotal);
    gemm_scaled_wmma<<<gridGemm, TPB, 0, stream>>>(x, normS, W1, BUF_A, NFEAT);
    scatter_add<<<gridEdgeWaves, TPB, 0, stream>>>(BUF_A, src, dst, BUF_B, E);
    bias_relu<<<gridFeat, TPB, 0, stream>>>(BUF_B, normD, b1, BUF_A, featTotal);

    // layer 2: M2 = (H1*normS)@W2 ; AGG2 = scatter
    gemm_scaled_wmma<<<gridGemm, TPB, 0, stream>>>(BUF_A, normS, W2, BUF_B, NH);
    zero_kernel<<<gridFeat, TPB, 0, stream>>>(BUF_A, featTotal);
    scatter_add<<<gridEdgeWaves, TPB, 0, stream>>>(BUF_B, src, dst, BUF_A, E);

    // final: out = (AGG2*normD + b2) @ Wfc + bfc
    final_fc<<<gridNodeWaves, TPB, 0, stream>>>(BUF_A, normD, b2, Wfc, bfc, out, N);
}